// Agent_88828513616344
// MI455X (gfx1250) — compile-verified
//
#include <hip/hip_runtime.h>
#include <hip/hip_bf16.h>
#include <math.h>

// ---------------------------------------------------------------------------
// Problem constants (from reference): N=4096 rows, F=8192 cols, D=208 features
//   irreps: 13x0e + 13x1o + 13x2e + 13x3o  (MUL=13, LMAX=3)
//   FAN_IN = 4*13*13 = 676
// ---------------------------------------------------------------------------
#define N_ROWS 4096
#define N_COLS 8192
#define KDIM   208
#define MUL    13

typedef float v2f __attribute__((ext_vector_type(2)));
typedef float v8f __attribute__((ext_vector_type(8)));

// ---------------------------------------------------------------------------
// Kernel 1: build Z[i,d] = c_l * sum_u x[i, off_l + u*m + k] * w_l[u, v]
//   layout within an l-block is u-major, m-minor (matches reshape(N,MUL,m))
//   output element d = off_l + v*m + k
// ---------------------------------------------------------------------------
__global__ __launch_bounds__(256) void Agent_88828513616344_prep(
    const float* __restrict__ x, const float* __restrict__ tp_w,
    float* __restrict__ Z)
{
    int gid = blockIdx.x * blockDim.x + threadIdx.x;
    if (gid >= N_ROWS * KDIM) return;
    int i = gid / KDIM;
    int d = gid % KDIM;

    int l, off;
    if (d < 13)       { l = 0; off = 0;   }
    else if (d < 52)  { l = 1; off = 13;  }
    else if (d < 117) { l = 2; off = 52;  }
    else              { l = 3; off = 117; }
    int m = 2 * l + 1;
    int r = d - off;
    int v = r / m;
    int k = r % m;

    float c = 1.0f / sqrtf((float)((2 * l + 1) * (4 * MUL * MUL)));
    const float* xr = x + (size_t)i * KDIM + off;
    const float* w  = tp_w + l * MUL * MUL;

    float s = 0.0f;
#pragma unroll
    for (int u = 0; u < MUL; ++u)
        s += xr[u * m + k] * w[u * MUL + v];

    Z[(size_t)i * KDIM + d] = c * s;
}

// ---------------------------------------------------------------------------
// Deterministic "gumbel" noise: murmur3-finalizer hash of the flat index,
// mixed with the fixed seed (42), mapped to (0,1), then -log(-log(u)).
// ---------------------------------------------------------------------------
__device__ __forceinline__ float gumbel_noise(unsigned int idx) {
    unsigned int h = idx * 0x9E3779B9u + 42u * 0x85EBCA6Bu;
    h ^= h >> 16; h *= 0x85EBCA6Bu;
    h ^= h >> 13; h *= 0xC2B2AE35u;
    h ^= h >> 16;
    float u = ((float)(h >> 8) + 0.5f) * (1.0f / 16777216.0f); // (0,1)
    return -__logf(-__logf(u));
}

// ---------------------------------------------------------------------------
// Kernel 2: fused  features = Z * Y^T  (f32 WMMA 16x16x4)  +  masked
// Gumbel-argmax over columns + value = features @ critic_w + critic_b.
//
// One block = 16 rows; 8 waves stripe the 512 column tiles, TWO tiles per
// wave iteration: one ds_load of the shared A fragment feeds two independent
// WMMA accumulator chains (halves LDS traffic, doubles loads in flight,
// breaks the single D->C dependency chain between consecutive WMMAs).
//
// WMMA f32 16x16x4 operand mapping (wave32):
//   A (16x4):  float2 at (row = lane&15, k + 2*(lane>>4))   [from LDS]
//   B (4x16):  float2 at (col = lane&15, k + 2*(lane>>4))   [from global]
//   C/D:       VGPR j -> row j (lanes 0-15) / j+8 (lanes 16-31), N = lane&15
// ---------------------------------------------------------------------------
#define A_STRIDE 210  // 208 padded -> distinct LDS banks for 16 rows

__global__ __launch_bounds__(256) void Agent_88828513616344_main(
    const float* __restrict__ Z, const float* __restrict__ Y,
    const unsigned char* __restrict__ masks,
    const float* __restrict__ critic_w, const float* __restrict__ critic_b,
    float* __restrict__ out)
{
    __shared__ float shA[16 * A_STRIDE];
    __shared__ float sKey[16][8];
    __shared__ int   sIdx[16][8];
    __shared__ float sVal[16][8];

    const int tid  = threadIdx.x;
    const int wid  = tid >> 5;
    const int lane = tid & 31;
    const int lo   = lane & 15;
    const int hi   = lane >> 4;            // 0 or 1
    const int rowbase = blockIdx.x * 16;

    // Stage the 16x208 A-tile into LDS.
    for (int idx = tid; idx < 16 * KDIM; idx += 256) {
        int r = idx / KDIM, c = idx % KDIM;
        shA[r * A_STRIDE + c] = Z[(size_t)(rowbase + r) * KDIM + c];
    }
    __syncthreads();

    float bk[8], vacc[8];
    int   bi[8];
#pragma unroll
    for (int j = 0; j < 8; ++j) { bk[j] = -INFINITY; bi[j] = 0; vacc[j] = 0.0f; }

    const float* aBase = &shA[lo * A_STRIDE + hi * 2];

    // 512 column tiles of 16; wave w owns tile pairs (2w, 2w+1), (2w+16, ...).
    for (int t = 2 * wid; t < (N_COLS / 16); t += 16) {
        const int cb0 = t << 4;            // first tile's column base
        const int cb1 = cb0 + 16;          // second tile's column base
        const float* b0Base = Y + (size_t)(cb0 + lo) * KDIM + hi * 2;
        const float* b1Base = Y + (size_t)(cb1 + lo) * KDIM + hi * 2;

        v8f acc0 = {};
        v8f acc1 = {};
#pragma unroll 4
        for (int k = 0; k < KDIM; k += 4) {
            v2f a  = *(const v2f*)(aBase + k);    // ds_load_b64 (shared A)
            v2f b0 = *(const v2f*)(b0Base + k);   // global_load_b64
            v2f b1 = *(const v2f*)(b1Base + k);   // global_load_b64
            acc0 = __builtin_amdgcn_wmma_f32_16x16x4_f32(
                false, a, false, b0, (short)0, acc0, false, false);
            acc1 = __builtin_amdgcn_wmma_f32_16x16x4_f32(
                false, a, false, b1, (short)0, acc1, false, false);
        }

        // Fused epilogue for both tiles: masked gumbel-argmax + value accum.
#pragma unroll
        for (int p = 0; p < 2; ++p) {
            const int   col = (p ? cb1 : cb0) + lo;
            const float cwv = critic_w[col];
#pragma unroll
            for (int j = 0; j < 8; ++j) {
                const int row = rowbase + j + (hi << 3);
                float feat = p ? acc1[j] : acc0[j];
                unsigned char mk = masks[(size_t)row * N_COLS + col];
                float logit = mk ? feat : -3.402823466e+38f;  // finfo(f32).min
                float key = logit + gumbel_noise((unsigned int)row * N_COLS + col);
                if (key > bk[j] || (key == bk[j] && col < bi[j])) {
                    bk[j] = key; bi[j] = col;
                }
                vacc[j] += feat * cwv;
            }
        }
    }

    // Reduce across the 16 lanes of each half-wave (rows j / j+8 independent).
#pragma unroll
    for (int j = 0; j < 8; ++j) {
#pragma unroll
        for (int s = 1; s < 16; s <<= 1) {
            float ok = __shfl_xor(bk[j],   s, 16);
            int   oi = __shfl_xor(bi[j],   s, 16);
            float ov = __shfl_xor(vacc[j], s, 16);
            if (ok > bk[j] || (ok == bk[j] && oi < bi[j])) { bk[j] = ok; bi[j] = oi; }
            vacc[j] += ov;
        }
    }

    // Lanes 0 and 16 hold per-row results; publish to LDS.
    if (lo == 0) {
#pragma unroll
        for (int j = 0; j < 8; ++j) {
            int row = j + (hi << 3);
            sKey[row][wid] = bk[j];
            sIdx[row][wid] = bi[j];
            sVal[row][wid] = vacc[j];
        }
    }
    __syncthreads();

    // Final reduce across the 8 waves; write actions (as float) and values.
    if (tid < 16) {
        float bkk = sKey[tid][0];
        int   bii = sIdx[tid][0];
        float vs  = sVal[tid][0];
#pragma unroll
        for (int w = 1; w < 8; ++w) {
            float ok = sKey[tid][w];
            int   oi = sIdx[tid][w];
            if (ok > bkk || (ok == bkk && oi < bii)) { bkk = ok; bii = oi; }
            vs += sVal[tid][w];
        }
        out[rowbase + tid]          = (float)bii;          // actions
        out[N_ROWS + rowbase + tid] = vs + critic_b[0];    // value
    }
}

// ---------------------------------------------------------------------------
// Launch: inputs are (x, fragment_environments, tp_w, critic_w, critic_b,
// masks) in setup_inputs() order. d_ws holds Z (4096*208 f32 = 3.4 MB).
// ---------------------------------------------------------------------------
extern "C" void kernel_launch(void* const* d_in, const int* in_sizes, int n_in,
                              void* d_out, int out_size, void* d_ws, size_t ws_size,
                              hipStream_t stream) {
    const float*         x     = (const float*)d_in[0];
    const float*         Y     = (const float*)d_in[1];
    const float*         tp_w  = (const float*)d_in[2];
    const float*         cw    = (const float*)d_in[3];
    const float*         cb    = (const float*)d_in[4];
    const unsigned char* masks = (const unsigned char*)d_in[5];

    float* Z   = (float*)d_ws;
    float* out = (float*)d_out;

    int prep_threads = N_ROWS * KDIM;
    Agent_88828513616344_prep<<<(prep_threads + 255) / 256, 256, 0, stream>>>(
        x, tp_w, Z);

    Agent_88828513616344_main<<<N_ROWS / 16, 256, 0, stream>>>(
        Z, Y, masks, cw, cb, out);
}